// KDTree_37744172597258
// MI455X (gfx1250) — compile-verified
//
#include <hip/hip_runtime.h>
#include <stdint.h>

typedef __attribute__((ext_vector_type(2))) float v2f;
typedef __attribute__((ext_vector_type(8))) float v8f;
typedef unsigned long long u64_t;

#define N        4096
#define DIMS     512
#define NTHREADS 1024
#define RPT      (N / NTHREADS)   // 4 rows per thread

// Monotone bijection float -> u32 (order-preserving for all finite values).
__device__ __forceinline__ unsigned fmap(float f) {
  unsigned u = __float_as_uint(f);
  return u ^ ((unsigned)((int)u >> 31) | 0x80000000u);
}

__global__ __launch_bounds__(NTHREADS)
void kdtree_select_kernel(const float* __restrict__ feat,
                          const float* __restrict__ qry,
                          float* __restrict__ out) {
  // Staging slots for (col0, col1) pairs; reused in-place as composite col-0 keys.
  __shared__ u64_t        s_key[N];     // 32 KB
  __shared__ float        s_k1[N];      // 16 KB
  __shared__ unsigned char s_side[N];   //  4 KB  (0=left, 1=right, 2=root)
  __shared__ int s_root, s_lc, s_rc, s_sel0, s_sel1;

  const int tid = threadIdx.x;
  const int b   = blockIdx.x;
  const float* fb = feat + (size_t)b * N * DIMS;
  const float* qb = qry  + (size_t)b * DIMS;

  // ---- Phase A: async gather of (col0,col1) of every row straight into LDS.
  // One global_load_async_to_lds_b64 per row; tracked by ASYNCcnt (no VGPR data).
  #pragma unroll
  for (int r = 0; r < RPT; ++r) {
    int i = tid + r * NTHREADS;
    u64_t   ga = (u64_t)(uintptr_t)(fb + (size_t)i * DIMS);
    unsigned la = (unsigned)(uintptr_t)&s_key[i];   // LDS byte offset (addrspace(3) low bits)
    asm volatile("global_load_async_to_lds_b64 %0, %1, off"
                 :: "v"(la), "v"(ga) : "memory");
  }
  asm volatile("s_wait_asynccnt 0x0" ::: "memory");
  __syncthreads();

  // ---- Phase B: build stable-sort composite keys for column 0 (in place).
  #pragma unroll
  for (int r = 0; r < RPT; ++r) {
    int i = tid + r * NTHREADS;
    u64_t v = s_key[i];
    float f0 = __uint_as_float((unsigned)v);
    float f1 = __uint_as_float((unsigned)(v >> 32));
    s_key[i] = ((u64_t)fmap(f0) << 12) | (unsigned)i;  // index tiebreak == stable sort
    s_k1[i]  = f1;
  }
  __syncthreads();

  // ---- Phase C: rank every row on column 0 by counting (LDS broadcast inner loop).
  u64_t myk[RPT]; int cnt[RPT];
  #pragma unroll
  for (int r = 0; r < RPT; ++r) { myk[r] = s_key[tid + r * NTHREADS]; cnt[r] = 0; }
  for (int j = 0; j < N; ++j) {
    u64_t kj = s_key[j];
    #pragma unroll
    for (int r = 0; r < RPT; ++r) cnt[r] += (kj < myk[r]) ? 1 : 0;
  }
  #pragma unroll
  for (int r = 0; r < RPT; ++r) {
    int i = tid + r * NTHREADS;
    int s = (cnt[r] < N / 2) ? 0 : ((cnt[r] == N / 2) ? 2 : 1);
    s_side[i] = (unsigned char)s;
    if (s == 2) s_root = i;                 // unique writer (ranks are a bijection)
  }
  __syncthreads();

  // ---- Phase D: conditional rank on column 1 within each half.
  u64_t myk1[RPT]; int side[RPT]; int cnt1[RPT];
  #pragma unroll
  for (int r = 0; r < RPT; ++r) {
    int i = tid + r * NTHREADS;
    side[r] = s_side[i];
    myk1[r] = ((u64_t)fmap(s_k1[i]) << 12) | (unsigned)i;
    cnt1[r] = 0;
  }
  for (int j = 0; j < N; ++j) {
    int   sj = s_side[j];
    u64_t kj = ((u64_t)fmap(s_k1[j]) << 12) | (unsigned)j;
    #pragma unroll
    for (int r = 0; r < RPT; ++r)
      cnt1[r] += (sj == side[r] && kj < myk1[r]) ? 1 : 0;
  }
  #pragma unroll
  for (int r = 0; r < RPT; ++r) {
    int i = tid + r * NTHREADS;
    if (side[r] == 0 && cnt1[r] == 1024) s_lc = i;  // left  half: 2048 elems, median idx 1024
    if (side[r] == 1 && cnt1[r] == 1023) s_rc = i;  // right half: 2047 elems, median idx 1023
  }
  __syncthreads();

  // ---- Phase E: 512-dim squared distances via V_WMMA_F32_16X16X4_F32 (wave 0, full EXEC).
  if (tid < 32) {
    const int lane = tid;
    const int root_i = s_root, lc = s_lc, rc = s_rc;
    const bool gl = qb[0] < fb[(size_t)root_i * DIMS];
    int cands[3];
    cands[0] = gl ? lc : rc;   // nxt
    cands[1] = root_i;         // root
    cands[2] = gl ? rc : lc;   // opp
    float d2[3];
    // A 16x4 f32 layout: lanes 0-15 hold K=0,1; lanes 16-31 hold K=2,3; M = lane%16.
    const int base = (lane & 15) * 4 + ((lane >> 4) & 1) * 2;
    for (int c = 0; c < 3; ++c) {
      const float* row = fb + (size_t)cands[c] * DIMS;
      v8f acc = {0.f, 0.f, 0.f, 0.f, 0.f, 0.f, 0.f, 0.f};
      #pragma unroll
      for (int g = 0; g < 8; ++g) {          // 8 groups x 64 elems = 512
        int e = g * 64 + base;
        float2 a  = *(const float2*)(row + e);
        float2 qv = *(const float2*)(qb  + e);
        v2f dv = { a.x - qv.x, a.y - qv.y };
        // D = diff * diff^T ; diagonal accumulates chunk self-dot-products.
        acc = __builtin_amdgcn_wmma_f32_16x16x4_f32(false, dv, false, dv,
                                                    (short)0, acc, false, false);
      }
      // Extract diagonal of 16x16 f32 C/D (VGPR r = row r (lanes 0-15) / row r+8 (lanes 16-31)).
      float part = 0.f;
      #pragma unroll
      for (int r = 0; r < 8; ++r) {
        int m = (lane >= 16) ? (r + 8) : r;
        if (m == (lane & 15)) part += acc[r];
      }
      #pragma unroll
      for (int off = 16; off; off >>= 1) part += __shfl_xor(part, off, 32);
      d2[c] = part;   // squared distance (sqrt is monotone; ordering identical)
    }
    if (lane == 0) {
      // Stable top-2 of 3 (ties keep cand order [nxt, root, opp]).
      int o0 = 0, o1 = 1, o2 = 2;
      if (d2[o1] < d2[o0]) { int t = o0; o0 = o1; o1 = t; }
      if (d2[o2] < d2[o1]) {
        int t = o1; o1 = o2; o2 = t;
        if (d2[o1] < d2[o0]) { int t2 = o0; o0 = o1; o1 = t2; }
      }
      s_sel0 = cands[o0];
      s_sel1 = cands[o1];
    }
  }
  __syncthreads();

  // ---- Phase F: emit the two selected 512-float rows (1024 threads, 1024 floats).
  {
    int rowsel = tid >> 9;
    int k      = tid & (DIMS - 1);
    int src    = rowsel ? s_sel1 : s_sel0;
    out[((size_t)b * 2 + rowsel) * DIMS + k] = fb[(size_t)src * DIMS + k];
  }
}

extern "C" void kernel_launch(void* const* d_in, const int* in_sizes, int n_in,
                              void* d_out, int out_size, void* d_ws, size_t ws_size,
                              hipStream_t stream) {
  const float* feat = (const float*)d_in[0];   // (B, 4096, 512) f32
  const float* qry  = (const float*)d_in[1];   // (B, 512)       f32
  float* out = (float*)d_out;                  // (B, 2, 512)    f32
  int batches = in_sizes[1] / DIMS;            // 64
  kdtree_select_kernel<<<dim3(batches), dim3(NTHREADS), 0, stream>>>(feat, qry, out);
}